// ThunderCore_46428596470330
// MI455X (gfx1250) — compile-verified
//
#include <hip/hip_runtime.h>
#include <hip/hip_bf16.h>

typedef float v2f __attribute__((ext_vector_type(2)));
typedef float v8f __attribute__((ext_vector_type(8)));

#define KC 8
#define LDA_S 12   // sA row stride in floats: 48B rows -> 16B-aligned b128 dests,
                   // m*12 mod 64 distinct for m=0..15, +2-row offset disjoint banks
#define LDB_S 72   // sB row stride in floats: 2 rows = 144 floats = 16 banks offset

__device__ __forceinline__ unsigned lds_addr32(const void* p) {
    // generic LDS pointer: high 32 bits = shared aperture, low 32 bits = LDS offset
    return (unsigned)(uintptr_t)p;
}
__device__ __forceinline__ void async_copy_b128(unsigned lds, const float* gsrc) {
    asm volatile("global_load_async_to_lds_b128 %0, %1, off"
                 :: "v"(lds), "v"(gsrc) : "memory");
}

// ---------------------------------------------------------------------------
// fp32 GEMM via V_WMMA_F32_16X16X4_F32 with async global->LDS double buffering.
// Block tile 256(M) x 64(N), 256 threads = 8 waves (wave32).
// Wave grid 4(M) x 2(N): each wave owns 64x32 = 4x2 16x16 subtiles.
// Per wave per KC=8 chunk: exactly 3 async b128 issues (2 for A, 1 for B).
// ---------------------------------------------------------------------------
__global__ __launch_bounds__(256) void gemm_f32_wmma(
    const float* __restrict__ A, int lda,
    const float* __restrict__ B, int ldb,
    const float* __restrict__ bias, int nbias,
    float* __restrict__ C, int ldc,
    int M, int N, int K)
{
    __shared__ float sA[2][256 * LDA_S];   // [m][k] (first 8 cols used)
    __shared__ float sB[2][KC * LDB_S];    // [k][n] (first 64 cols used)

    const int tid  = threadIdx.x;
    const int lane = tid & 31;
    const int wave = tid >> 5;
    const int wm   = wave & 3;        // M group of 64 rows
    const int wn   = wave >> 2;       // N group of 32 cols
    const int row0 = blockIdx.y * 256;
    const int col0 = blockIdx.x * 64;
    const int half = lane >> 4;
    const int l16  = lane & 15;

    // Fixed staging coordinates: A row am handled by this thread (2 x b128),
    // B row (= wave id), 16B segment per lane 0..15 (1 x b128 per wave).
    const int am = (wave << 5) | lane;                    // 0..255
    const float* aptr = A + (long)(row0 + am) * lda;      // + k0
    const int bk = wave;                                  // 0..7
    const int bn = l16 << 2;
    const int gn = col0 + bn;
    const float* bptr = B + (long)bk * ldb + gn;          // + k0*ldb

    v8f acc[4][2];
    #pragma unroll
    for (int i = 0; i < 4; ++i)
        #pragma unroll
        for (int j = 0; j < 2; ++j)
            acc[i][j] = (v8f){0.f,0.f,0.f,0.f,0.f,0.f,0.f,0.f};

    const int nch = K / KC;   // K is a multiple of 8 for both GEMMs

    // Prologue: stage chunk 0 into buffer 0
    {
        unsigned la = lds_addr32(&sA[0][am * LDA_S]);
        async_copy_b128(la,      aptr);
        async_copy_b128(la + 16, aptr + 4);
        if (lane < 16) {
            if (gn < N) {
                async_copy_b128(lds_addr32(&sB[0][bk * LDB_S + bn]), bptr);
            } else {
                float* z = &sB[0][bk * LDB_S + bn];
                z[0] = 0.f; z[1] = 0.f; z[2] = 0.f; z[3] = 0.f;
            }
        }
    }

    for (int c = 0; c < nch; ++c) {
        const int cb = c & 1;
        if (c + 1 < nch) {
            // Stage next chunk into the other buffer (overlaps with compute)
            const int k0 = (c + 1) * KC;
            unsigned la = lds_addr32(&sA[cb ^ 1][am * LDA_S]);
            async_copy_b128(la,      aptr + k0);
            async_copy_b128(la + 16, aptr + k0 + 4);
            if (lane < 16) {
                if (gn < N) {
                    async_copy_b128(lds_addr32(&sB[cb ^ 1][bk * LDB_S + bn]),
                                    bptr + (long)k0 * ldb);
                } else {
                    float* z = &sB[cb ^ 1][bk * LDB_S + bn];
                    z[0] = 0.f; z[1] = 0.f; z[2] = 0.f; z[3] = 0.f;
                }
            }
            asm volatile("s_wait_asynccnt 3" ::: "memory");  // chunk c complete
        } else {
            asm volatile("s_wait_asynccnt 0" ::: "memory");
        }
        __syncthreads();

        const float* sAc = sA[cb];
        const float* sBc = sB[cb];
        #pragma unroll
        for (int kk = 0; kk < KC; kk += 4) {
            v2f af[4], bf[2];
            #pragma unroll
            for (int i = 0; i < 4; ++i) {
                int m = (wm << 6) + (i << 4) + l16;
                const float* p = &sAc[m * LDA_S + kk + (half << 1)];
                af[i].x = p[0];
                af[i].y = p[1];
            }
            #pragma unroll
            for (int j = 0; j < 2; ++j) {
                int n = (wn << 5) + (j << 4) + l16;
                bf[j].x = sBc[(kk + (half << 1) + 0) * LDB_S + n];
                bf[j].y = sBc[(kk + (half << 1) + 1) * LDB_S + n];
            }
            #pragma unroll
            for (int i = 0; i < 4; ++i)
                #pragma unroll
                for (int j = 0; j < 2; ++j)
                    acc[i][j] = __builtin_amdgcn_wmma_f32_16x16x4_f32(
                        false, af[i], false, bf[j],
                        (short)0, acc[i][j], false, false);
        }
        __syncthreads();
    }

    // Store: C/D layout — lane<16: M=v, lane>=16: M=v+8; N = lane%16
    #pragma unroll
    for (int i = 0; i < 4; ++i) {
        int rbase = row0 + (wm << 6) + (i << 4) + (half << 3);
        #pragma unroll
        for (int j = 0; j < 2; ++j) {
            int cidx = col0 + (wn << 5) + (j << 4) + l16;
            float bv = (cidx < nbias) ? bias[cidx] : 0.f;
            #pragma unroll
            for (int v = 0; v < 8; ++v)
                C[(long)(rbase + v) * ldc + cidx] = acc[i][j][v] + bv;
        }
    }
}

// ---------------------------------------------------------------------------
// Gate: comb[m, 1032+d] = sigmoid( comb[m, 0:1032] . W_g[:, d] + b_g[d] )
// One wave per row, 8 rows per block. W_g (1032x8 = 33KB) staged in LDS.
// ---------------------------------------------------------------------------
__global__ __launch_bounds__(256) void gate_kernel(
    float* __restrict__ comb, int ldc,
    const float* __restrict__ Wg, const float* __restrict__ bg)
{
    __shared__ float sWg[1032 * 8];
    const int tid = threadIdx.x;
    for (int idx = tid; idx < 1032 * 8; idx += 256) sWg[idx] = Wg[idx];
    __syncthreads();

    const int lane = tid & 31;
    const int wave = tid >> 5;
    const long row = (long)blockIdx.x * 8 + wave;

    float acc[8] = {0.f, 0.f, 0.f, 0.f, 0.f, 0.f, 0.f, 0.f};
    for (int k = lane; k < 1032; k += 32) {
        float xv = comb[row * ldc + k];
        #pragma unroll
        for (int d = 0; d < 8; ++d) acc[d] += xv * sWg[k * 8 + d];
    }
    #pragma unroll
    for (int off = 16; off > 0; off >>= 1)
        #pragma unroll
        for (int d = 0; d < 8; ++d)
            acc[d] += __shfl_xor(acc[d], off, 32);

    if (lane < 8) {
        float val = acc[lane] + bg[lane];
        val = 1.f / (1.f + __expf(-val));
        comb[row * ldc + 1032 + lane] = val;
    }
}

// ---------------------------------------------------------------------------
// Parallel affine scan: state_s = g_s * state_{s-1} + (1-g_s)*si_s.
// One block per (b,d) sequence (64 blocks). 256 threads x 16 steps each.
// ---------------------------------------------------------------------------
__global__ __launch_bounds__(256) void scan_kernel(
    float* __restrict__ comb, int ldc, float* __restrict__ final_state)
{
    __shared__ float sAc[256];
    __shared__ float sBc[256];
    const int t = threadIdx.x;
    const int b = blockIdx.x >> 3;
    const int d = blockIdx.x & 7;
    const int CH = 16;                       // 4096 / 256
    const long base = (long)b * 4096 * ldc;
    const int s0 = t * CH;

    float g[CH], si[CH];
    #pragma unroll
    for (int i = 0; i < CH; ++i) {
        long r = base + (long)(s0 + i) * ldc;
        g[i]  = comb[r + 1032 + d];
        si[i] = comb[r + 1024 + d];
    }

    float Ac = 1.f, Bc = 0.f;
    #pragma unroll
    for (int i = 0; i < CH; ++i) {
        float nA = g[i] * Ac;
        Bc = g[i] * Bc + (1.f - g[i]) * si[i];
        Ac = nA;
    }
    sAc[t] = Ac; sBc[t] = Bc;
    __syncthreads();

    for (int off = 1; off < 256; off <<= 1) {
        float pA = 1.f, pB = 0.f;
        if (t >= off) { pA = sAc[t - off]; pB = sBc[t - off]; }
        __syncthreads();
        if (t >= off) {
            float nA = sAc[t] * pA;
            float nB = sAc[t] * pB + sBc[t];
            sAc[t] = nA; sBc[t] = nB;
        }
        __syncthreads();
    }

    float state = (t == 0) ? 0.f : sBc[t - 1];

    #pragma unroll
    for (int i = 0; i < CH; ++i) {
        state = g[i] * state + (1.f - g[i]) * si[i];
        long r = base + (long)(s0 + i) * ldc;
        comb[r + 1024 + d] = state;
    }
    if (t == 255) final_state[b * 8 + d] = state;
}

// ---------------------------------------------------------------------------
extern "C" void kernel_launch(void* const* d_in, const int* in_sizes, int n_in,
                              void* d_out, int out_size, void* d_ws, size_t ws_size,
                              hipStream_t stream) {
    const float* x     = (const float*)d_in[0];   // (8, 4096, 1024)
    const float* W_in  = (const float*)d_in[1];   // (1024, 1032)
    const float* b_in  = (const float*)d_in[2];   // (1032,)
    const float* W_g   = (const float*)d_in[3];   // (1032, 8)
    const float* b_g   = (const float*)d_in[4];   // (8,)
    const float* W_out = (const float*)d_in[5];   // (1032, 1024)
    const float* b_out = (const float*)d_in[6];   // (1024,)

    float* out    = (float*)d_out;                    // (32768, 1024) flat
    float* fstate = out + (long)32768 * 1024;         // (8, 8) flat
    float* comb   = (float*)d_ws;                     // (32768, 1088) padded

    const int M = 32768;
    const int LD = 1088;                              // 17 * 64 (padded 1032)

    // Stage 1: combined = x @ W_in + b_in
    dim3 g1(LD / 64, M / 256);
    gemm_f32_wmma<<<g1, 256, 0, stream>>>(x, 1024, W_in, 1032, b_in, 1032,
                                          comb, LD, M, 1032, 1024);

    // Stage 2a: gate -> comb cols [1032, 1040)
    gate_kernel<<<M / 8, 256, 0, stream>>>(comb, LD, W_g, b_g);

    // Stage 2b: affine scan, states overwrite comb cols [1024, 1032)
    scan_kernel<<<64, 256, 0, stream>>>(comb, LD, fstate);

    // Stage 3: out = comb[:, :1032] @ W_out + b_out
    dim3 g3(1024 / 64, M / 256);
    gemm_f32_wmma<<<g3, 256, 0, stream>>>(comb, LD, W_out, 1024, b_out, 1024,
                                          out, 1024, M, 1024, 1032);
}